// ChannelTimeAttention_50242527429176
// MI455X (gfx1250) — compile-verified
//
#include <hip/hip_runtime.h>
#include <hip/hip_bf16.h>

#define BB 8
#define TT 16
#define CC 64
#define HH 56
#define WW 56
#define HWN (HH * WW)      // 3136
#define NTILES (HWN / 16)  // 196
#define DIN 64             // 8*8 pooled
#define DOUT 32            // 2*d_t

typedef __attribute__((ext_vector_type(2))) float v2f;
typedef __attribute__((ext_vector_type(8))) float v8f;

// ---------------------------------------------------------------------------
// Kernel 1: per n = b*64+c : 7x7 avg pool -> flat[16][64], q/k GEMMs, softmax
// att[16][16] written to workspace. Regular-temporal x reads (prime L2 for K2).
// ---------------------------------------------------------------------------
__global__ __launch_bounds__(256) void ctattn_pool_qk_att(
    const float* __restrict__ x, const float* __restrict__ Wq,
    const float* __restrict__ bq, const float* __restrict__ Wk,
    const float* __restrict__ bk, float* __restrict__ att_g) {
  __shared__ float flat[TT][DIN];   // 4 KB
  __shared__ float qs[TT][DOUT];    // 2 KB
  __shared__ float ks[TT][DOUT];    // 2 KB
  __shared__ float att[TT][TT];     // 1 KB

  const int n = blockIdx.x;             // n = b*CC + c
  const int bb = n >> 6, cc = n & 63;
  const int tid = threadIdx.x;

  // ---- adaptive avg pool (56 -> 8, i.e. 7x7 blocks): 1024 bins, 4/thread
  const float inv49 = 1.0f / 49.0f;
#pragma unroll
  for (int i = 0; i < 4; ++i) {
    int bin = tid + i * 256;
    int t = bin >> 6, p = bin & 63;
    int py = p >> 3, px = p & 7;
    const float* src =
        x + ((((size_t)bb * TT + t) * CC + cc) * HH + (size_t)py * 7) * WW + px * 7;
    float s = 0.f;
    for (int dy = 0; dy < 7; ++dy) {
      const float* row = src + (size_t)dy * WW;
#pragma unroll
      for (int dx = 0; dx < 7; ++dx) s += row[dx];
    }
    flat[t][p] = s * inv49;
  }
  __syncthreads();

  // ---- q = flat@Wq + bq, k = flat@Wk + bk ; 512 elems each, 2 per thread
#pragma unroll
  for (int i = 0; i < 2; ++i) {
    int e = tid + i * 256;
    int t = e >> 5, o = e & 31;
    float aq = bq[o], ak = bk[o];
    for (int p = 0; p < DIN; ++p) {
      float f = flat[t][p];
      aq = fmaf(f, Wq[p * DOUT + o], aq);
      ak = fmaf(f, Wk[p * DOUT + o], ak);
    }
    qs[t][o] = aq;
    ks[t][o] = ak;
  }
  __syncthreads();

  // ---- logits att[t][s] = dot(q[t], k[s]) / sqrt(16)
  {
    int t = tid >> 4, s = tid & 15;
    float a = 0.f;
    for (int o = 0; o < DOUT; ++o) a = fmaf(qs[t][o], ks[s][o], a);
    att[t][s] = a * 0.25f;
  }
  __syncthreads();

  // ---- softmax over s, write to workspace
  {
    int t = tid >> 4, s = tid & 15;
    float m = att[t][0];
    for (int i = 1; i < TT; ++i) m = fmaxf(m, att[t][i]);
    float e = __expf(att[t][s] - m);
    __syncthreads();
    att[t][s] = e;
    __syncthreads();
    float sum = 0.f;
    for (int i = 0; i < TT; ++i) sum += att[t][i];
    att_g[(size_t)n * 256 + t * 16 + s] = e / sum;
  }
}

// ---------------------------------------------------------------------------
// Kernel 2: out[n] = att[n] (16x16) @ v[n] (16x3136) via
// V_WMMA_F32_16X16X4_F32 (4 chained K-slices per 16x16 output tile).
// Grid (512, 2), 8 waves/block; wave handles tiles (by*8+w), step 16.
// x reads are last-use -> nontemporal; out stores are streaming -> nontemporal.
// ---------------------------------------------------------------------------
__global__ __launch_bounds__(256) void ctattn_apply(
    const float* __restrict__ x, const float* __restrict__ att_g,
    float* __restrict__ out) {
  const int n = blockIdx.x;
  const int bb = n >> 6, cc = n & 63;
  const int lane = threadIdx.x & 31;
  const int wave = (blockIdx.y << 3) + (threadIdx.x >> 5);  // 0..15

  const float* attn = att_g + (size_t)n * 256;
  const float* vb = x + ((size_t)bb * TT * CC + cc) * (size_t)HWN;
  float* ob = out + ((size_t)bb * TT * CC + cc) * (size_t)HWN;
  const size_t SSTR = (size_t)CC * HWN;   // stride between s-rows of v: 200704

  const int m = lane & 15;      // A row (M) / B,D column (N) index for this lane
  const int hi = lane >> 4;     // lane-half selects K pair (A,B) / M+8 (D)

  // A slices (att) in 16x4 f32 A-layout: VGPR0 = K{0|2}, VGPR1 = K{1|3}
  v2f a[4];
#pragma unroll
  for (int w = 0; w < 4; ++w) {
    int kb = 4 * w + (hi << 1);
    v2f av;
    av.x = attn[m * 16 + kb];
    av.y = attn[m * 16 + kb + 1];
    a[w] = av;
  }

#pragma unroll 2
  for (int tile = wave; tile < NTILES; tile += 16) {   // uniform per wave
    const int j = tile * 16;
    v8f acc = {};
#pragma unroll
    for (int w = 0; w < 4; ++w) {
      int kb = 4 * w + (hi << 1);
      v2f bv;
      bv.x = __builtin_nontemporal_load(&vb[(size_t)kb * SSTR + j + m]);
      bv.y = __builtin_nontemporal_load(&vb[(size_t)(kb + 1) * SSTR + j + m]);
      acc = __builtin_amdgcn_wmma_f32_16x16x4_f32(
          /*neg_a=*/false, a[w], /*neg_b=*/false, bv,
          /*c_mod=*/(short)0, acc, /*reuse_a=*/false, /*reuse_b=*/false);
    }
    // D layout: VGPR r -> M = r (lanes 0-15) / M = r+8 (lanes 16-31), N = m
#pragma unroll
    for (int r = 0; r < 8; ++r) {
      int mm = r + (hi << 3);
      __builtin_nontemporal_store(acc[r], &ob[(size_t)mm * SSTR + j + m]);
    }
  }
}

extern "C" void kernel_launch(void* const* d_in, const int* in_sizes, int n_in,
                              void* d_out, int out_size, void* d_ws, size_t ws_size,
                              hipStream_t stream) {
  const float* x  = (const float*)d_in[0];
  const float* Wq = (const float*)d_in[1];
  const float* bq = (const float*)d_in[2];
  const float* Wk = (const float*)d_in[3];
  const float* bk = (const float*)d_in[4];
  float* out = (float*)d_out;
  float* att_g = (float*)d_ws;   // 512 * 256 floats = 512 KB

  ctattn_pool_qk_att<<<dim3(BB * CC), 256, 0, stream>>>(x, Wq, bq, Wk, bk, att_g);
  ctattn_apply<<<dim3(BB * CC, 2), 256, 0, stream>>>(x, att_g, out);
}